// STARModel_59846074303041
// MI455X (gfx1250) — compile-verified
//
#include <hip/hip_runtime.h>
#include <hip/hip_bf16.h>

// STAR/SMPL LBS pipeline for MI455X (gfx1250).
// Bandwidth-bound: ~590 MB of fp32 traffic (posedirs = 518 MB dominates),
// ~0.43 GFLOP -> roofline ~25us @ 23.3 TB/s. Keep fp32 everywhere; use
// V_WMMA_F32_16X16X4_F32 for the weights(V,24) @ [rot|trans](24,12) blend
// (exact fp32, 6 K=4 steps per 16-vertex tile, wave32).

typedef float v2f __attribute__((ext_vector_type(2)));
typedef float v8f __attribute__((ext_vector_type(8)));

#define KJ    24
#define NBETA 10

// Workspace layout (float offsets)
#define OFF_PF   0      // 216  pose_feat = (R - I)
#define OFF_R    256    // 216  rotation matrices
#define OFF_JACC 512    // 72   joint-position accumulator
#define OFF_T    640    // 384  padded 24x16 transform table [rot(9)|trans(3)|0000]
#define OFF_VSH  1024   // V*3  v_shaped

// ---------------------------------------------------------------------------
// Kernel 1: Rodrigues per joint + zero J accumulator.
// ---------------------------------------------------------------------------
__global__ __launch_bounds__(128) void star_init_kernel(
    const float* __restrict__ pose, float* __restrict__ R,
    float* __restrict__ pf, float* __restrict__ Jacc) {
  int tid = threadIdx.x;
  if (tid < 72) Jacc[tid] = 0.0f;
  if (tid < KJ) {
    float rx = pose[tid * 3 + 0], ry = pose[tid * 3 + 1], rz = pose[tid * 3 + 2];
    float ex = rx + 1e-12f, ey = ry + 1e-12f, ez = rz + 1e-12f;
    float theta = sqrtf(ex * ex + ey * ey + ez * ez);
    float inv = 1.0f / theta;
    float kx = rx * inv, ky = ry * inv, kz = rz * inv;
    float s = sinf(theta), c = cosf(theta), oc = 1.0f - c;
    float Km[9] = {0.f, -kz, ky, kz, 0.f, -kx, -ky, kx, 0.f};
    float K2[9];
#pragma unroll
    for (int i = 0; i < 3; ++i)
#pragma unroll
      for (int j = 0; j < 3; ++j) {
        float a = 0.f;
#pragma unroll
        for (int q = 0; q < 3; ++q) a += Km[i * 3 + q] * Km[q * 3 + j];
        K2[i * 3 + j] = a;
      }
#pragma unroll
    for (int n = 0; n < 9; ++n) {
      float off = s * Km[n] + oc * K2[n];
      float eye = ((n == 0) || (n == 4) || (n == 8)) ? 1.0f : 0.0f;
      R[tid * 9 + n]  = eye + off;
      pf[tid * 9 + n] = off;   // R - I
    }
  }
}

// ---------------------------------------------------------------------------
// Kernel 2: v_shaped = v_template + shapedirs*betas ; J = J_regressor@v_shaped
// (wave shfl reduction -> LDS -> global atomics).
// ---------------------------------------------------------------------------
__global__ __launch_bounds__(256) void star_shape_kernel(
    const float* __restrict__ vt, const float* __restrict__ sd,
    const float* __restrict__ betas, const float* __restrict__ Jreg,
    float* __restrict__ vsh, float* __restrict__ Jacc, int V) {
  __shared__ float sB[NBETA];
  __shared__ float sJ[72];
  int tid = threadIdx.x;
  if (tid < NBETA) sB[tid] = betas[tid];
  if (tid < 72) sJ[tid] = 0.0f;
  __syncthreads();

  int v = blockIdx.x * 256 + tid;
  int vc = v < V ? v : V - 1;
  bool valid = v < V;

  const float* sdp = sd + (size_t)vc * 30;
  float vx = vt[vc * 3 + 0], vy = vt[vc * 3 + 1], vz = vt[vc * 3 + 2];
#pragma unroll
  for (int i = 0; i < NBETA; ++i) {
    float b = sB[i];
    vx += sdp[i] * b;
    vy += sdp[10 + i] * b;
    vz += sdp[20 + i] * b;
  }
  if (valid) {
    vsh[vc * 3 + 0] = vx;
    vsh[vc * 3 + 1] = vy;
    vsh[vc * 3 + 2] = vz;
  }

  int lane = tid & 31;
#pragma unroll 4
  for (int k = 0; k < KJ; ++k) {
    float w = valid ? Jreg[(size_t)k * V + v] : 0.0f;
    float cx = w * vx, cy = w * vy, cz = w * vz;
#pragma unroll
    for (int off = 16; off > 0; off >>= 1) {
      cx += __shfl_xor(cx, off);
      cy += __shfl_xor(cy, off);
      cz += __shfl_xor(cz, off);
    }
    if (lane == 0) {
      atomicAdd(&sJ[k * 3 + 0], cx);
      atomicAdd(&sJ[k * 3 + 1], cy);
      atomicAdd(&sJ[k * 3 + 2], cz);
    }
  }
  __syncthreads();
  if (tid < 72) atomicAdd(&Jacc[tid], sJ[tid]);
}

// ---------------------------------------------------------------------------
// Kernel 3: serial 24-joint kinematic chain -> padded 24x16 transform table.
// ---------------------------------------------------------------------------
__global__ void star_chain_kernel(const float* __restrict__ R,
                                  const float* __restrict__ Jacc,
                                  float* __restrict__ T) {
  if (threadIdx.x != 0 || blockIdx.x != 0) return;
  const int PAR[KJ] = {0, 0, 0, 0, 1, 2, 3, 4, 5, 6, 7, 8,
                       9, 9, 9, 12, 13, 14, 16, 17, 18, 19, 20, 21};
  float J[KJ][3], rel[KJ][3], Ar[KJ][9], At[KJ][3];
  for (int k = 0; k < KJ; ++k)
    for (int c = 0; c < 3; ++c) J[k][c] = Jacc[k * 3 + c];
  for (int c = 0; c < 3; ++c) rel[0][c] = J[0][c];
  for (int j = 1; j < KJ; ++j)
    for (int c = 0; c < 3; ++c) rel[j][c] = J[j][c] - J[PAR[j]][c];
  for (int n = 0; n < 9; ++n) Ar[0][n] = R[n];
  for (int c = 0; c < 3; ++c) At[0][c] = rel[0][c];
  for (int j = 1; j < KJ; ++j) {
    int p = PAR[j];
    for (int i = 0; i < 3; ++i)
      for (int c = 0; c < 3; ++c) {
        float a = 0.f;
        for (int q = 0; q < 3; ++q) a += Ar[p][i * 3 + q] * R[j * 9 + q * 3 + c];
        Ar[j][i * 3 + c] = a;
      }
    for (int i = 0; i < 3; ++i) {
      float a = At[p][i];
      for (int q = 0; q < 3; ++q) a += Ar[p][i * 3 + q] * rel[j][q];
      At[j][i] = a;
    }
  }
  for (int j = 0; j < KJ; ++j) {
    for (int n = 0; n < 9; ++n) T[j * 16 + n] = Ar[j][n];
    for (int i = 0; i < 3; ++i) {
      float tr = At[j][i];
      for (int q = 0; q < 3; ++q) tr -= Ar[j][i * 3 + q] * J[j][q];
      T[j * 16 + 9 + i] = tr;
    }
    T[j * 16 + 12] = 0.f;
    T[j * 16 + 13] = 0.f;
    T[j * 16 + 14] = 0.f;
    T[j * 16 + 15] = 0.f;
  }
}

// ---------------------------------------------------------------------------
// Kernel 4: posedirs matvec stream (518 MB) + WMMA f32 skinning blend.
// Each wave32 owns 32 vertices = two 16x16x4-WMMA tiles (M=16,N=16(12),K=24).
// All WMMAs run with full EXEC (no divergent branches around them).
// ---------------------------------------------------------------------------
__global__ __launch_bounds__(256) void star_skin_kernel(
    const float* __restrict__ posedirs, const float* __restrict__ weights,
    const float* __restrict__ vsh, const float* __restrict__ pf,
    const float* __restrict__ T, float* __restrict__ out, int V) {
  __shared__ float4 sPF[54];        // pose_feat as float4
  __shared__ float sT[KJ * 16];     // padded transform table
  __shared__ float sC[8 * 2 * 256]; // per-wave WMMA C tiles (16 KB)
  int tid = threadIdx.x;
  if (tid < 54) sPF[tid] = ((const float4*)pf)[tid];
  for (int i = tid; i < KJ * 16; i += 256) sT[i] = T[i];
  __syncthreads();

  int wave = tid >> 5;
  int lane = tid & 31;
  int m = lane & 15;   // row within 16-vertex tile / B column n
  int hi = lane >> 4;  // lane half
  int base = blockIdx.x * 256 + wave * 32;
  int v = base + lane;
  int vc = v < V ? v : V - 1;

  // --- pose-corrective matvec: 648 floats/vertex streamed as b128 loads ---
  const float4* pd = (const float4*)(posedirs + (size_t)vc * 648);
  float ax = 0.f, ay = 0.f, az = 0.f;
#pragma unroll 3
  for (int p = 0; p < 54; ++p) {
    float4 f = sPF[p];
    float4 x = pd[p];
    float4 y = pd[p + 54];
    float4 z = pd[p + 108];
    ax += x.x * f.x + x.y * f.y + x.z * f.z + x.w * f.w;
    ay += y.x * f.x + y.y * f.y + y.z * f.z + y.w * f.w;
    az += z.x * f.x + z.y * f.y + z.z * f.z + z.w * f.w;
  }
  float vpx = vsh[vc * 3 + 0] + ax;
  float vpy = vsh[vc * 3 + 1] + ay;
  float vpz = vsh[vc * 3 + 2] + az;

  // --- blended transforms: C[16x16] = weights[16x24] @ T[24x16] via WMMA ---
  // A 16x4 f32 layout: v0 = K(0|2), v1 = K(1|3) split on lane halves.
  // B 4x16 f32 layout: v0 = rows K(0|2), v1 = rows K(1|3), n = lane&15.
  int koff = hi ? 2 : 0;
#pragma unroll
  for (int t = 0; t < 2; ++t) {
    int tb = base + t * 16;
    int vm = tb + m;
    vm = vm < V ? vm : V - 1;  // clamp loads; WMMA keeps full EXEC
    const float* wrow = weights + (size_t)vm * KJ;
    v8f acc = {0.f, 0.f, 0.f, 0.f, 0.f, 0.f, 0.f, 0.f};
#pragma unroll
    for (int ck = 0; ck < 6; ++ck) {
      v2f a, b;
      a.x = wrow[ck * 4 + koff + 0];
      a.y = wrow[ck * 4 + koff + 1];
      b.x = sT[(ck * 4 + koff + 0) * 16 + m];
      b.y = sT[(ck * 4 + koff + 1) * 16 + m];
      acc = __builtin_amdgcn_wmma_f32_16x16x4_f32(false, a, false, b,
                                                  (short)0, acc, false, false);
    }
    // C layout: VGPR r -> row (r + 8*hi), col (lane&15); stage through LDS.
    float* cdst = &sC[(wave * 2 + t) * 256];
#pragma unroll
    for (int r = 0; r < 8; ++r) cdst[(r + hi * 8) * 16 + m] = acc[r];
  }
  __syncthreads();

  // --- apply blended 3x4 transform to v_posed ---
  if (v < V) {
    const float* e = &sC[(wave * 2 + hi) * 256 + m * 16];
    out[v * 3 + 0] = e[0] * vpx + e[1] * vpy + e[2] * vpz + e[9];
    out[v * 3 + 1] = e[3] * vpx + e[4] * vpy + e[5] * vpz + e[10];
    out[v * 3 + 2] = e[6] * vpx + e[7] * vpy + e[8] * vpz + e[11];
  }
}

// ---------------------------------------------------------------------------
extern "C" void kernel_launch(void* const* d_in, const int* in_sizes, int n_in,
                              void* d_out, int out_size, void* d_ws,
                              size_t ws_size, hipStream_t stream) {
  const float* betas      = (const float*)d_in[0];
  const float* pose       = (const float*)d_in[1];
  const float* v_template = (const float*)d_in[2];
  const float* shapedirs  = (const float*)d_in[3];
  const float* posedirs   = (const float*)d_in[4];
  const float* Jreg       = (const float*)d_in[5];
  const float* weights    = (const float*)d_in[6];
  float* out = (float*)d_out;

  float* ws   = (float*)d_ws;
  float* pf   = ws + OFF_PF;
  float* R    = ws + OFF_R;
  float* Jacc = ws + OFF_JACC;
  float* T    = ws + OFF_T;
  float* vsh  = ws + OFF_VSH;

  int V = in_sizes[2] / 3;  // 200000
  int nb = (V + 255) / 256;

  star_init_kernel<<<1, 128, 0, stream>>>(pose, R, pf, Jacc);
  star_shape_kernel<<<nb, 256, 0, stream>>>(v_template, shapedirs, betas, Jreg,
                                            vsh, Jacc, V);
  star_chain_kernel<<<1, 32, 0, stream>>>(R, Jacc, T);
  star_skin_kernel<<<nb, 256, 0, stream>>>(posedirs, weights, vsh, pf, T, out,
                                           V);
}